// SAGPooling_54778012893228
// MI455X (gfx1250) — compile-verified
//
#include <hip/hip_runtime.h>
#include <hip/hip_bf16.h>

#define N_NODES 100000
#define N_GRAPHS 50
#define NODES_PER_GRAPH 2000
#define N_EDGES 1600000
#define D_FEAT 128
#define K_SEL 1000
#define X_OUT_ELEMS (N_GRAPHS * K_SEL * D_FEAT)   // 6,400,000

typedef __attribute__((ext_vector_type(2))) float v2f;
typedef __attribute__((ext_vector_type(8))) float v8f;

// ---------------------------------------------------------------------------
// Kernel 1: zero the per-node edge-aggregate scalar buffer t[N_NODES]
// ---------------------------------------------------------------------------
__global__ void zero_kernel(float* __restrict__ t, int n) {
    int i = blockIdx.x * blockDim.x + threadIdx.x;
    if (i < n) t[i] = 0.0f;
}

// ---------------------------------------------------------------------------
// Kernel 2: per-node dots s_l = x.W_l, s_r = x.W_r via V_WMMA_F32_16X16X4_F32.
// One wave handles a 16-node tile. A = 16x4 f32 chunk of x (ISA layout:
// lanes 0-15 -> K=k,k+1 ; lanes 16-31 -> K=k+2,k+3). B column 0 = W_l chunk,
// column 1 = W_r chunk, other columns zeroed BRANCHLESSLY via a per-lane
// multiplicative mask (previous version used a null-pointer branch that
// compiled to s_and_saveexec around every wmma step).
// C/D layout: element (M,N) lives in VGPR M%8, lane N + 16*(M>=8): lane 0
// holds s_l for M=0..7, lane 16 s_l for M=8..15, lanes 1/17 same for s_r.
// ---------------------------------------------------------------------------
__global__ void dots_kernel(const float* __restrict__ x,
                            const float* __restrict__ Wl,
                            const float* __restrict__ Wr,
                            float* __restrict__ sl,
                            float* __restrict__ sr) {
    const int wave = (blockIdx.x * blockDim.x + threadIdx.x) >> 5;
    const int lane = threadIdx.x & 31;
    if (wave >= (N_NODES / 16)) return;          // wave-uniform: EXEC stays all-1s

    const int node  = wave * 16 + (lane & 15);
    const int khalf = (lane >> 4) << 1;          // 0 for lanes 0-15, 2 for 16-31
    const int ncol  = lane & 15;                 // B/N column owned by this lane

    // Branchless B source: every lane loads from a valid 512B weight array
    // (broadcast, L0-resident); columns >= 2 are zeroed by the mask.
    const float* wsrc = (ncol == 1) ? Wr : Wl;
    const float  msk  = (ncol < 2) ? 1.0f : 0.0f;

    const float* xrow = x + (size_t)node * D_FEAT;
    v8f c = {};
#pragma unroll
    for (int k = 0; k < D_FEAT; k += 4) {
        const int klo = k + khalf;
        v2f a = *(const v2f*)(xrow + klo);       // global_load_b64
        v2f w = *(const v2f*)(wsrc + klo);       // global_load_b64 (cached)
        v2f b;
        b.x = w.x * msk;
        b.y = w.y * msk;
        c = __builtin_amdgcn_wmma_f32_16x16x4_f32(
                /*neg_a=*/false, a, /*neg_b=*/false, b,
                /*c_mod=*/(short)0, c, /*reuse_a=*/false, /*reuse_b=*/false);
    }

    const int base = wave * 16;
    if (lane == 0)  { for (int j = 0; j < 8; ++j) sl[base + j]     = c[j]; }
    if (lane == 16) { for (int j = 0; j < 8; ++j) sl[base + 8 + j] = c[j]; }
    if (lane == 1)  { for (int j = 0; j < 8; ++j) sr[base + j]     = c[j]; }
    if (lane == 17) { for (int j = 0; j < 8; ++j) sr[base + 8 + j] = c[j]; }
}

// ---------------------------------------------------------------------------
// Kernel 3: edge scatter of SCALARS: t[col] += s_l[row]. 4B atomic per edge
// instead of 512B row RMW (the linear layer commutes with the neighbor sum).
// t is 0.4 MB -> L2-resident atomics.
// ---------------------------------------------------------------------------
__global__ void scatter_kernel(const long long* __restrict__ ei,
                               const float* __restrict__ sl,
                               float* __restrict__ t) {
    int i = blockIdx.x * blockDim.x + threadIdx.x;
    if (i < N_EDGES) {
        int r = (int)ei[i];            // edge_index[0, i]
        int c = (int)ei[N_EDGES + i];  // edge_index[1, i]
        atomicAdd(&t[c], sl[r]);
    }
}

// ---------------------------------------------------------------------------
// Kernel 4: per-graph top-K via 2048-entry bitonic sort in LDS on
// (score desc, index asc) -- matches jax.lax.top_k tie ordering.
// Score is computed on load: tanh(t + s_r + b). 50 blocks x 1024 threads.
// ---------------------------------------------------------------------------
__global__ __launch_bounds__(1024)
void topk_kernel(const float* __restrict__ t,
                 const float* __restrict__ sr,
                 const float* __restrict__ bl,
                 int* __restrict__ perm,
                 float* __restrict__ sel) {
    __shared__ float s[2048];
    __shared__ int   id[2048];
    const int g = blockIdx.x;
    const float b = bl[0];

    for (int i = threadIdx.x; i < 2048; i += 1024) {
        if (i < NODES_PER_GRAPH) {
            int n = g * NODES_PER_GRAPH + i;
            s[i]  = tanhf(t[n] + sr[n] + b);
            id[i] = i;
        } else {
            s[i]  = -__builtin_inff();
            id[i] = 0x3FFFFFFF;
        }
    }
    __syncthreads();

    for (int k = 2; k <= 2048; k <<= 1) {
        for (int j = k >> 1; j > 0; j >>= 1) {
            for (int i = threadIdx.x; i < 2048; i += 1024) {
                int ixj = i ^ j;
                if (ixj > i) {
                    bool up = ((i & k) == 0);   // block sorted in desired order
                    float si = s[i], sj = s[ixj];
                    int   ii = id[i], ij = id[ixj];
                    bool good = (si > sj) || (si == sj && ii < ij);
                    if (good != up) {
                        s[i] = sj; s[ixj] = si;
                        id[i] = ij; id[ixj] = ii;
                    }
                }
            }
            __syncthreads();
        }
    }

    if (threadIdx.x < K_SEL) {
        int o = g * K_SEL + threadIdx.x;
        perm[o] = g * NODES_PER_GRAPH + id[threadIdx.x];
        sel[o]  = s[threadIdx.x];
    }
}

// ---------------------------------------------------------------------------
// Kernel 5: gather + scale selected rows; one wave per output row (float4
// per lane => global_load_b128 / global_store_b128). Also emits batch_out.
// ---------------------------------------------------------------------------
__global__ void gather_kernel(const float* __restrict__ x,
                              const long long* __restrict__ batch,
                              const int* __restrict__ perm,
                              const float* __restrict__ sel,
                              float* __restrict__ out,
                              int batch_as_i64) {
    const int row  = (blockIdx.x * blockDim.x + threadIdx.x) >> 5;
    const int lane = threadIdx.x & 31;
    if (row >= N_GRAPHS * K_SEL) return;

    const int   p  = perm[row];
    const float sc = sel[row];
    const float4* src = (const float4*)(x + (size_t)p * D_FEAT);
    float4*       dst = (float4*)(out + (size_t)row * D_FEAT);
    float4 v = src[lane];
    v.x *= sc; v.y *= sc; v.z *= sc; v.w *= sc;
    dst[lane] = v;

    if (lane == 0) {
        long long bv = batch[p];
        if (batch_as_i64) {
            ((long long*)(out + X_OUT_ELEMS))[row] = bv;
        } else {
            out[X_OUT_ELEMS + row] = (float)bv;
        }
    }
}

// ---------------------------------------------------------------------------
// Host launcher
// ---------------------------------------------------------------------------
extern "C" void kernel_launch(void* const* d_in, const int* in_sizes, int n_in,
                              void* d_out, int out_size, void* d_ws, size_t ws_size,
                              hipStream_t stream) {
    const float*     x     = (const float*)d_in[0];
    const long long* ei    = (const long long*)d_in[1];
    const long long* batch = (const long long*)d_in[2];
    const float*     Wl    = (const float*)d_in[3];
    const float*     bl    = (const float*)d_in[4];
    const float*     Wr    = (const float*)d_in[5];
    float*           out   = (float*)d_out;

    // Workspace layout (256B aligned slices)
    char* ws = (char*)d_ws;
    float* t    = (float*)(ws + 0);         // N_NODES floats   (400,000 B)
    float* sl   = (float*)(ws + 409600);    // N_NODES floats
    float* sr   = (float*)(ws + 819200);    // N_NODES floats
    int*   perm = (int*)  (ws + 1228800);   // 50,000 ints
    float* sel  = (float*)(ws + 1433600);   // 50,000 floats

    const int batch_as_i64 = (out_size - X_OUT_ELEMS) >= 2 * N_GRAPHS * K_SEL;

    // 1. zero the scalar aggregate
    zero_kernel<<<(N_NODES + 255) / 256, 256, 0, stream>>>(t, N_NODES);

    // 2. per-node dot products via WMMA (6250 tiles, 8 waves/block)
    {
        int waves  = N_NODES / 16;               // 6250
        int blocks = (waves + 7) / 8;            // 782
        dots_kernel<<<blocks, 256, 0, stream>>>(x, Wl, Wr, sl, sr);
    }

    // 3. scalar edge scatter
    scatter_kernel<<<(N_EDGES + 255) / 256, 256, 0, stream>>>(ei, sl, t);

    // 4. per-graph top-K (fused score + tanh)
    topk_kernel<<<N_GRAPHS, 1024, 0, stream>>>(t, sr, bl, perm, sel);

    // 5. gather/scale output rows + batch_out
    {
        int rows   = N_GRAPHS * K_SEL;           // 50,000 (one wave each)
        int blocks = (rows + 7) / 8;             // 6250
        gather_kernel<<<blocks, 256, 0, stream>>>(x, batch, perm, sel, out,
                                                  batch_as_i64);
    }
}